// DecoderRNN_9852654977473
// MI455X (gfx1250) — compile-verified
//
#include <hip/hip_runtime.h>
#include <hip/hip_bf16.h>
#include <math.h>

// Problem constants (match reference).
#define BB 32
#define PP 196
#define DD 2048
#define AA 512
#define EE 512
#define HH 512
#define VV 10000
#define LL 21
#define TT 20

typedef float v2f __attribute__((ext_vector_type(2)));
typedef float v8f __attribute__((ext_vector_type(8)));

__device__ __forceinline__ float sigmoidf_(float x) { return 1.f / (1.f + __expf(-x)); }

// ---------------------------------------------------------------------------
// Setup: stable descending argsort of lengths (B=32), gather caps, emit
// sort_ind / dec_lens / caps outputs (cast to float for the float output buf).
// ---------------------------------------------------------------------------
__global__ void __launch_bounds__(BB)
setup_kernel(const int* __restrict__ lengths, const int* __restrict__ caps_in,
             int* __restrict__ sind, int* __restrict__ dlen, int* __restrict__ caps_s,
             float* __restrict__ out_caps, float* __restrict__ out_dlen,
             float* __restrict__ out_sind) {
    int i = threadIdx.x;
    int li = lengths[i];
    int rank = 0;
    for (int j = 0; j < BB; ++j) {
        int lj = lengths[j];
        if (lj > li || (lj == li && j < i)) rank++;
    }
    sind[rank] = i;
    __syncthreads();
    int src = sind[i];
    int l = lengths[src];
    dlen[i] = l - 1;
    out_dlen[i] = (float)(l - 1);
    out_sind[i] = (float)src;
    for (int t = 0; t < LL; ++t) {
        int cv = caps_in[src * LL + t];
        caps_s[i * LL + t] = cv;
        out_caps[i * LL + t] = (float)cv;
    }
}

// ---------------------------------------------------------------------------
// mean over P of (sorted) encoder_out -> mean_enc[B,D]
// ---------------------------------------------------------------------------
__global__ void mean_kernel(const float* __restrict__ enc, const int* __restrict__ sind,
                            float* __restrict__ mean) {
    int idx = blockIdx.x * blockDim.x + threadIdx.x;
    if (idx >= BB * DD) return;
    int b = idx / DD, d = idx % DD;
    const float* p = enc + ((size_t)sind[b] * PP) * DD + d;
    float s = 0.f;
    for (int j = 0; j < PP; ++j) s += p[(size_t)j * DD];
    mean[idx] = s * (1.0f / PP);
}

// ---------------------------------------------------------------------------
// f32 WMMA GEMM, 16(M) x (16*NSUB)(N) tile per wave32, V_WMMA_F32_16X16X4_F32.
// NSUB is a compile-time constant -> completely branch-free inner loop:
// one b64 A-fragment load feeds NSUB independent WMMAs per K-step.
//   A fragment (16x4, ISA 7.12.2): lane<16 -> M=lane, K={k0,k0+1};
//                                  lane>=16 -> M=lane-16, K={k0+2,k0+3}
//   B fragment (4x16): v0: row K=k0+2*half, col n&15 ; v1: next K row
//   C/D: vgpr r -> row m0 + r + 8*(lane>>4), col nsub + (lane&15)
// rowmap: if sind != null, A row of output-row m is sind[m/rpb]*rpb + m%rpb.
// declens/tstep: zero-fill rows where !(tstep < declens[row]).
// nbase: column offset of this launch (for N-remainder splits).
// ---------------------------------------------------------------------------
template <int NSUB>
__global__ void __launch_bounds__(32)
wmma_gemm_f32(const float* __restrict__ A, int lda,
              const float* __restrict__ Bm,
              const float* __restrict__ bias,
              float* __restrict__ C, int ldc,
              int N, int K, int nbase,
              const int* __restrict__ sind, int rpb,
              int accumulate,
              const int* __restrict__ declens, int tstep) {
    const int lane = threadIdx.x & 31;
    const int half = lane >> 4;
    const int l16  = lane & 15;
    const int m0 = blockIdx.y << 4;
    const int n0 = nbase + blockIdx.x * (16 * NSUB);

    int am = m0 + l16;
    int arow = am;
    if (sind) arow = sind[am / rpb] * rpb + (am % rpb);
    const float* Arow = A + (size_t)arow * lda;

    v8f acc[NSUB];
#pragma unroll
    for (int j = 0; j < NSUB; ++j)
        acc[j] = (v8f){0.f, 0.f, 0.f, 0.f, 0.f, 0.f, 0.f, 0.f};
    if (accumulate) {
#pragma unroll
        for (int j = 0; j < NSUB; ++j)
#pragma unroll
            for (int r = 0; r < 8; ++r)
                acc[j][r] = C[(size_t)(m0 + r + 8 * half) * ldc + n0 + 16 * j + l16];
    }

    // One-shot prefetch of the B panel head (global_prefetch_b8).
    __builtin_prefetch(Bm + (size_t)(2 * half) * N + n0 + l16 * 4, 0, 1);

    for (int k0 = 0; k0 < K; k0 += 4) {
        // A fragment: two consecutive f32 -> single b64 load (8B aligned).
        v2f a = *(const v2f*)(Arow + k0 + 2 * half);

        const float* Brow0 = Bm + (size_t)(k0 + 2 * half) * N;
        const float* Brow1 = Brow0 + N;

        v2f b[NSUB];
#pragma unroll
        for (int j = 0; j < NSUB; ++j) {
            int col = n0 + 16 * j + l16;
            b[j].x = Brow0[col];
            b[j].y = Brow1[col];
        }
#pragma unroll
        for (int j = 0; j < NSUB; ++j)
            acc[j] = __builtin_amdgcn_wmma_f32_16x16x4_f32(
                false, a, false, b[j], (short)0, acc[j], false, false);
    }

#pragma unroll
    for (int j = 0; j < NSUB; ++j) {
#pragma unroll
        for (int r = 0; r < 8; ++r) {
            int row = m0 + r + 8 * half;
            int col = n0 + 16 * j + l16;
            float v = acc[j][r] + (bias ? bias[col] : 0.f);
            if (declens && !(tstep < declens[row])) v = 0.f;
            C[(size_t)row * ldc + col] = v;
        }
    }
}

// ---------------------------------------------------------------------------
// attention energy: e[b,p] = full_att_b + sum_a relu(att1[b*P+p,a]+att2[b,a])*fw[a]
// one block (128 threads) per (b,p) row
// ---------------------------------------------------------------------------
__global__ void __launch_bounds__(128)
att_e_kernel(const float* __restrict__ att1, const float* __restrict__ att2,
             const float* __restrict__ fw, const float* __restrict__ fb,
             float* __restrict__ eout) {
    int bp = blockIdx.x;
    int b = bp / PP;
    const float* a1 = att1 + (size_t)bp * AA;
    const float* a2 = att2 + (size_t)b * AA;
    float s = 0.f;
    for (int a = threadIdx.x; a < AA; a += 128) {
        float v = a1[a] + a2[a];
        v = v > 0.f ? v : 0.f;
        s += v * fw[a];
    }
    __shared__ float red[128];
    red[threadIdx.x] = s;
    __syncthreads();
    for (int st = 64; st > 0; st >>= 1) {
        if (threadIdx.x < st) red[threadIdx.x] += red[threadIdx.x + st];
        __syncthreads();
    }
    if (threadIdx.x == 0) eout[bp] = red[0] + fb[0];
}

// ---------------------------------------------------------------------------
// softmax over P (in-place e -> alpha); also writes masked alpha to output
// ---------------------------------------------------------------------------
__global__ void __launch_bounds__(256)
softmax_kernel(float* __restrict__ e, float* __restrict__ alphas_out,
               const int* __restrict__ declens, int t) {
    int b = blockIdx.x;
    float* eb = e + (size_t)b * PP;
    __shared__ float red[256];
    int tid = threadIdx.x;

    float m = -1e30f;
    for (int p = tid; p < PP; p += 256) m = fmaxf(m, eb[p]);
    red[tid] = m;
    __syncthreads();
    for (int st = 128; st > 0; st >>= 1) {
        if (tid < st) red[tid] = fmaxf(red[tid], red[tid + st]);
        __syncthreads();
    }
    m = red[0];
    __syncthreads();

    float s = 0.f;
    for (int p = tid; p < PP; p += 256) {
        float v = __expf(eb[p] - m);
        eb[p] = v;
        s += v;
    }
    red[tid] = s;
    __syncthreads();
    for (int st = 128; st > 0; st >>= 1) {
        if (tid < st) red[tid] += red[tid + st];
        __syncthreads();
    }
    float inv = 1.f / red[0];
    int active = (t < declens[b]) ? 1 : 0;
    for (int p = tid; p < PP; p += 256) {
        float a = eb[p] * inv;
        eb[p] = a;
        alphas_out[(size_t)b * TT * PP + (size_t)t * PP + p] = active ? a : 0.f;
    }
}

// ---------------------------------------------------------------------------
// awe[b,d] = sum_p alpha[b,p] * enc[sind[b],p,d]
// ---------------------------------------------------------------------------
__global__ void awe_kernel(const float* __restrict__ enc, const int* __restrict__ sind,
                           const float* __restrict__ alpha, float* __restrict__ awe) {
    int idx = blockIdx.x * blockDim.x + threadIdx.x;
    if (idx >= BB * DD) return;
    int b = idx / DD, d = idx % DD;
    const float* ep = enc + ((size_t)sind[b] * PP) * DD + d;
    const float* al = alpha + (size_t)b * PP;
    float s = 0.f;
    for (int p = 0; p < PP; ++p) s += al[p] * ep[(size_t)p * DD];
    awe[idx] = s;
}

// ---------------------------------------------------------------------------
// x[b, 0:E] = emb[caps_s[b,t]] ; x[b, E:E+D] = sigmoid(gpre[b,d]) * awe[b,d]
// ---------------------------------------------------------------------------
__global__ void xbuild_kernel(const float* __restrict__ emb, const int* __restrict__ caps_s,
                              int t, const float* __restrict__ gpre,
                              const float* __restrict__ awe, float* __restrict__ x) {
    int idx = blockIdx.x * blockDim.x + threadIdx.x;
    const int XW = EE + DD;
    if (idx >= BB * XW) return;
    int b = idx / XW, j = idx % XW;
    float v;
    if (j < EE) {
        v = emb[(size_t)caps_s[b * LL + t] * EE + j];
    } else {
        int d = j - EE;
        v = sigmoidf_(gpre[(size_t)b * DD + d]) * awe[(size_t)b * DD + d];
    }
    x[idx] = v;
}

// ---------------------------------------------------------------------------
// LSTM cell elementwise; update h,c only where t < dec_lens[b]
// g layout per row: [i | f | g | o], each H wide
// ---------------------------------------------------------------------------
__global__ void cell_kernel(const float* __restrict__ g, float* __restrict__ h,
                            float* __restrict__ c, const int* __restrict__ declens, int t) {
    int idx = blockIdx.x * blockDim.x + threadIdx.x;
    if (idx >= BB * HH) return;
    int b = idx / HH, j = idx % HH;
    const float* gb = g + (size_t)b * 4 * HH;
    float ig = sigmoidf_(gb[j]);
    float fg = sigmoidf_(gb[HH + j]);
    float gg = tanhf(gb[2 * HH + j]);
    float og = sigmoidf_(gb[3 * HH + j]);
    float cn = fg * c[idx] + ig * gg;
    float hn = og * tanhf(cn);
    if (t < declens[b]) {
        c[idx] = cn;
        h[idx] = hn;
    }
}

// ---------------------------------------------------------------------------
// Host launcher
// ---------------------------------------------------------------------------
extern "C" void kernel_launch(void* const* d_in, const int* in_sizes, int n_in,
                              void* d_out, int out_size, void* d_ws, size_t ws_size,
                              hipStream_t stream) {
    const float* encoder_out = (const float*)d_in[0];
    const int*   enc_caps    = (const int*)d_in[1];
    const int*   cap_lens    = (const int*)d_in[2];
    const float* enc_att_w   = (const float*)d_in[3];
    const float* enc_att_b   = (const float*)d_in[4];
    const float* dec_att_w   = (const float*)d_in[5];
    const float* dec_att_b   = (const float*)d_in[6];
    const float* full_att_w  = (const float*)d_in[7];
    const float* full_att_b  = (const float*)d_in[8];
    const float* emb         = (const float*)d_in[9];
    const float* init_h_w    = (const float*)d_in[10];
    const float* init_h_b    = (const float*)d_in[11];
    const float* init_c_w    = (const float*)d_in[12];
    const float* init_c_b    = (const float*)d_in[13];
    const float* f_beta_w    = (const float*)d_in[14];
    const float* f_beta_b    = (const float*)d_in[15];
    const float* lstm_w_ih   = (const float*)d_in[16];
    const float* lstm_w_hh   = (const float*)d_in[17];
    const float* lstm_b      = (const float*)d_in[18];
    const float* fc_w        = (const float*)d_in[19];
    const float* fc_b        = (const float*)d_in[20];

    // Output layout (floats, concatenated in return order):
    // predictions (B*T*V) | caps (B*L) | dec_lens (B) | alphas (B*T*P) | sort_ind (B)
    float* out      = (float*)d_out;
    float* out_pred = out;
    float* out_caps = out + (size_t)BB * TT * VV;
    float* out_dlen = out_caps + (size_t)BB * LL;
    float* out_alph = out_dlen + BB;
    float* out_sind = out_alph + (size_t)BB * TT * PP;

    // Workspace carving (float units)
    float* wsf   = (float*)d_ws;
    int*   sind  = (int*)d_ws;              // 32
    int*   dlen  = sind + BB;               // 32
    int*   capsS = dlen + BB;               // 32*21 = 672
    float* mean  = wsf + 1024;              // B*D   = 65536
    float* h     = mean + (size_t)BB * DD;  // B*H
    float* c     = h + (size_t)BB * HH;     // B*H
    float* att2  = c + (size_t)BB * HH;     // B*A
    float* alpha = att2 + (size_t)BB * AA;  // B*P (padded to 8192)
    float* gpre  = alpha + 8192;            // B*D
    float* awe   = gpre + (size_t)BB * DD;  // B*D
    float* x     = awe + (size_t)BB * DD;   // B*(E+D) = 81920
    float* g     = x + (size_t)BB * (EE + DD);   // B*4H
    float* att1  = g + (size_t)BB * 4 * HH;      // (B*P)*A = 3,211,264

    const int NT64_A  = AA / 64;         // 8
    const int NT64_H  = HH / 64;         // 8
    const int NT64_D  = DD / 64;         // 32
    const int NT64_4H = (4 * HH) / 64;   // 32
    const int NT64_V  = VV / 64;         // 156 full tiles; +16-col remainder

    // --- setup: sort, caps gather, int outputs ---
    setup_kernel<<<1, BB, 0, stream>>>(cap_lens, enc_caps, sind, dlen, capsS,
                                       out_caps, out_dlen, out_sind);

    // --- mean encoder ---
    mean_kernel<<<(BB * DD + 255) / 256, 256, 0, stream>>>(encoder_out, sind, mean);

    // --- h0, c0 (M=32, N=512, K=2048) ---
    wmma_gemm_f32<4><<<dim3(NT64_H, BB / 16), 32, 0, stream>>>(
        mean, DD, init_h_w, init_h_b, h, HH, HH, DD, 0, nullptr, 1, 0, nullptr, 0);
    wmma_gemm_f32<4><<<dim3(NT64_H, BB / 16), 32, 0, stream>>>(
        mean, DD, init_c_w, init_c_b, c, HH, HH, DD, 0, nullptr, 1, 0, nullptr, 0);

    // --- att1 = enc_sorted @ enc_att_w + b   (M=6272, N=512, K=2048) ---
    wmma_gemm_f32<4><<<dim3(NT64_A, (BB * PP) / 16), 32, 0, stream>>>(
        encoder_out, DD, enc_att_w, enc_att_b, att1, AA, AA, DD, 0, sind, PP, 0, nullptr, 0);

    // --- timestep loop (fixed sequence; graph-capture friendly) ---
    for (int t = 0; t < TT; ++t) {
        // att2 = h @ dec_att_w + b   (32x512x512)
        wmma_gemm_f32<4><<<dim3(NT64_A, BB / 16), 32, 0, stream>>>(
            h, HH, dec_att_w, dec_att_b, att2, AA, AA, HH, 0, nullptr, 1, 0, nullptr, 0);

        // e[b,p] = relu(att1 + att2) . full_att_w + b
        att_e_kernel<<<BB * PP, 128, 0, stream>>>(att1, att2, full_att_w, full_att_b, alpha);

        // softmax over P -> alpha; masked alpha to output
        softmax_kernel<<<BB, 256, 0, stream>>>(alpha, out_alph, dlen, t);

        // awe[b,d] = sum_p alpha * enc
        awe_kernel<<<(BB * DD + 255) / 256, 256, 0, stream>>>(encoder_out, sind, alpha, awe);

        // gate pre-activation: gpre = h @ f_beta_w + b   (32x2048x512)
        wmma_gemm_f32<4><<<dim3(NT64_D, BB / 16), 32, 0, stream>>>(
            h, HH, f_beta_w, f_beta_b, gpre, DD, DD, HH, 0, nullptr, 1, 0, nullptr, 0);

        // x = [emb_t, sigmoid(gpre)*awe]
        xbuild_kernel<<<(BB * (EE + DD) + 255) / 256, 256, 0, stream>>>(
            emb, capsS, t, gpre, awe, x);

        // g = x @ W_ih + lstm_b   (32x2048x2560)
        wmma_gemm_f32<4><<<dim3(NT64_4H, BB / 16), 32, 0, stream>>>(
            x, EE + DD, lstm_w_ih, lstm_b, g, 4 * HH, 4 * HH, EE + DD, 0,
            nullptr, 1, 0, nullptr, 0);
        // g += h @ W_hh            (32x2048x512, accumulate)
        wmma_gemm_f32<4><<<dim3(NT64_4H, BB / 16), 32, 0, stream>>>(
            h, HH, lstm_w_hh, nullptr, g, 4 * HH, 4 * HH, HH, 0,
            nullptr, 1, 1, nullptr, 0);

        // LSTM cell, masked update of h,c
        cell_kernel<<<(BB * HH + 255) / 256, 256, 0, stream>>>(g, h, c, dlen, t);

        // preds = h @ fc_w + fc_b, masked-zero, strided store into (B,T,V)
        float* predt = out_pred + (size_t)t * VV;
        wmma_gemm_f32<4><<<dim3(NT64_V, BB / 16), 32, 0, stream>>>(
            h, HH, fc_w, fc_b, predt, TT * VV, VV, HH, 0, nullptr, 1, 0, dlen, t);
        // FC remainder: last 16 columns (9984..9999), NSUB=1 specialization
        wmma_gemm_f32<1><<<dim3(1, BB / 16), 32, 0, stream>>>(
            h, HH, fc_w, fc_b, predt, TT * VV, VV, HH, NT64_V * 64,
            nullptr, 1, 0, dlen, t);
    }
}